// Fourier_Edge_extractor_3358664426354
// MI455X (gfx1250) — compile-verified
//
#include <hip/hip_runtime.h>
#include <hip/hip_bf16.h>
#include <math.h>

// ---------------------------------------------------------------------------
// Problem constants (match reference)
// ---------------------------------------------------------------------------
#define BB 32
#define CC 64
#define HH 128
#define WW 128
#define HW (HH * WW)              // 16384
#define MTOT ((size_t)BB * HW)    // 524288 pixels
#define NTOT (MTOT * CC)          // 33,554,432 floats per tensor

// FFT scratchpad: 128 rows x 129-pitch of float2 (pitch avoids bank conflicts
// on the column pass: stride 129*2 dwords -> bank advances by 2 per row).
#define FPITCH 129
#define FFT_LDS_BYTES (128 * FPITCH * (int)sizeof(float2))   // 132096 B (needs CDNA5 320KB LDS)

typedef float v2f __attribute__((ext_vector_type(2)));
typedef float v8f __attribute__((ext_vector_type(8)));

// ---------------------------------------------------------------------------
// Kernel 1: per-(b,c) image: FFT2 -> radial high-pass -> IFFT2 -> |.|
// Output written channel-last: xh[b][hw][c]
// ---------------------------------------------------------------------------
__device__ __forceinline__ void bitrev_elems(float2* sm, int lineStride, int elemStride) {
  for (int i = threadIdx.x; i < 128 * 128; i += 256) {
    int line = i >> 7;
    int p    = i & 127;
    int bp   = (int)(__brev((unsigned)p) >> 25);   // 7-bit bit reverse
    if (p < bp) {
      float2* a = sm + line * lineStride + p  * elemStride;
      float2* b = sm + line * lineStride + bp * elemStride;
      float2 t = *a; *a = *b; *b = t;
    }
  }
  __syncthreads();
}

__device__ __forceinline__ void fft_pass(float2* sm, int lineStride, int elemStride, float sign) {
  // 128 independent length-128 radix-2 DIT FFTs (input bit-reversed).
  for (int s = 0; s < 7; ++s) {
    const int half = 1 << s;
    for (int t = threadIdx.x; t < 128 * 64; t += 256) {
      int line = t >> 6;
      int idx  = t & 63;
      int pos  = idx & (half - 1);
      int k    = ((idx >> s) << (s + 1)) + pos;
      float ang = sign * 3.14159265358979f * (float)pos / (float)half;
      float sw, cw;
      __sincosf(ang, &sw, &cw);
      float2* pa = sm + line * lineStride + k * elemStride;
      float2* pb = pa + half * elemStride;
      float2 a = *pa, b = *pb;
      float2 wb = make_float2(cw * b.x - sw * b.y, cw * b.y + sw * b.x);
      *pa = make_float2(a.x + wb.x, a.y + wb.y);
      *pb = make_float2(a.x - wb.x, a.y - wb.y);
    }
    __syncthreads();
  }
}

__global__ __launch_bounds__(256) void fft_highpass_kernel(
    const float* __restrict__ x, float* __restrict__ xh, const int* __restrict__ radius_p) {
  extern __shared__ float2 sm[];
  const int img  = blockIdx.x;          // b*C + c
  const int bidx = img >> 6;
  const int cidx = img & 63;
  const float* src = x + (size_t)img * HW;
  const int r2 = radius_p[0] * radius_p[0];

  for (int i = threadIdx.x; i < HW; i += 256) {
    int r = i >> 7, c = i & 127;
    sm[r * FPITCH + c] = make_float2(src[i], 0.0f);
  }
  __syncthreads();

  // Forward 2-D FFT
  bitrev_elems(sm, FPITCH, 1);  fft_pass(sm, FPITCH, 1, -1.0f);   // rows
  bitrev_elems(sm, 1, FPITCH);  fft_pass(sm, 1, FPITCH, -1.0f);   // cols

  // High-pass: fftshift + (dis<radius -> zero) + ifftshift collapses to
  // zeroing coefficients whose signed-frequency radius^2 < r^2.
  for (int i = threadIdx.x; i < HW; i += 256) {
    int u = i >> 7, v = i & 127;
    int fu = (u < 64) ? u : u - 128;
    int fv = (v < 64) ? v : v - 128;
    if (fu * fu + fv * fv < r2) sm[u * FPITCH + v] = make_float2(0.0f, 0.0f);
  }
  __syncthreads();

  // Inverse 2-D FFT (sign flipped), scale 1/(H*W), magnitude
  bitrev_elems(sm, FPITCH, 1);  fft_pass(sm, FPITCH, 1, +1.0f);
  bitrev_elems(sm, 1, FPITCH);  fft_pass(sm, 1, FPITCH, +1.0f);

  const float scl = 1.0f / 16384.0f;
  float* dst = xh + (size_t)bidx * ((size_t)HW * CC) + cidx;
  for (int i = threadIdx.x; i < HW; i += 256) {
    int r = i >> 7, c = i & 127;
    float2 v = sm[r * FPITCH + c];
    dst[(size_t)i * CC] = sqrtf(v.x * v.x + v.y * v.y) * scl;
  }
}

// ---------------------------------------------------------------------------
// Kernel 2: depthwise ks x ks dilated conv + folded BN + ReLU (channel-last)
// ---------------------------------------------------------------------------
__global__ __launch_bounds__(256) void dw_aff_relu_kernel(
    const float* __restrict__ in, const float* __restrict__ w,
    const float* __restrict__ s, const float* __restrict__ bia,
    float* __restrict__ out, int ks, int dil, int pad) {
  size_t gid = (size_t)blockIdx.x * 256 + threadIdx.x;   // over B*HW*C
  int    c   = (int)(gid & 63);
  size_t p   = gid >> 6;                                 // b*HW + hw
  int    hw  = (int)(p & (HW - 1));
  size_t ib  = (p - (size_t)hw) * CC;                    // image base (floats)
  int h = hw >> 7, wd = hw & 127;

  float acc = 0.0f;
  for (int kh = 0; kh < ks; ++kh) {
    int ih = h - pad + kh * dil;
    if (ih < 0 || ih >= HH) continue;
    for (int kw = 0; kw < ks; ++kw) {
      int iw = wd - pad + kw * dil;
      if (iw < 0 || iw >= WW) continue;
      acc += w[(c * ks + kh) * ks + kw] * in[ib + (size_t)(ih * WW + iw) * CC + c];
    }
  }
  float v = s[c] * acc + bia[c];
  out[gid] = v > 0.0f ? v : 0.0f;
}

// ---------------------------------------------------------------------------
// Kernel 3: pointwise GEMM via V_WMMA_F32_16X16X4_F32 + affine (+ReLU) (+skip)
//   out[m][outOffset+o] = act( s[o]*sum_k in[m][k]*W[o][k] + b[o] ) (+ skip)
// Block: 256 threads = 8 waves; wave -> 16-row M tile; NTILES 16-wide N tiles.
// NTILES is a compile-time constant so acc[] stays in fixed VGPRs (no movrel).
// ---------------------------------------------------------------------------
template <int NTILES>
__global__ __launch_bounds__(256) void pw_wmma_kernel(
    const float* __restrict__ in, const float* __restrict__ W,
    const float* __restrict__ s, const float* __restrict__ bia,
    const float* __restrict__ skip, float* __restrict__ out,
    int outOffset, int doRelu) {
  constexpr int Cout = NTILES * 16;
  __shared__ float sW[Cout * 65];     // weights, 65-pitch (bank-conflict free)
  __shared__ float sA[8][16 * 65];    // per-wave A tile 16x64, 65-pitch

  const int tid  = threadIdx.x;
  const int wave = tid >> 5;
  const int lane = tid & 31;

  for (int i = tid; i < Cout * 64; i += 256)
    sW[(i >> 6) * 65 + (i & 63)] = W[i];

  const size_t mBase = (size_t)blockIdx.x * 128 + (size_t)wave * 16;
  for (int i = lane; i < 16 * 64; i += 32) {
    int mr = i >> 6, k = i & 63;
    sA[wave][mr * 65 + k] = in[(mBase + mr) * CC + k];
  }
  __syncthreads();

  // fp32 WMMA fragment indexing (ISA 7.12.2):
  //  A 16x4: lanes 0-15 -> K={k0,k0+1}, lanes 16-31 -> K={k0+2,k0+3}, M=lane&15
  //  B 4x16: same K split, N=lane&15
  //  C/D 16x16: VGPR v -> M=v (lanes 0-15) / v+8 (lanes 16-31), N=lane&15
  const int mrow  = lane & 15;
  const int khalf = (lane >> 4) * 2;

  v8f acc[NTILES];
  const v8f zero = {0.f, 0.f, 0.f, 0.f, 0.f, 0.f, 0.f, 0.f};
#pragma unroll
  for (int nt = 0; nt < NTILES; ++nt) acc[nt] = zero;

#pragma unroll
  for (int k0 = 0; k0 < 64; k0 += 4) {
    v2f a;
    a.x = sA[wave][mrow * 65 + k0 + khalf];
    a.y = sA[wave][mrow * 65 + k0 + khalf + 1];
#pragma unroll
    for (int nt = 0; nt < NTILES; ++nt) {
      int n = nt * 16 + mrow;
      v2f b;
      b.x = sW[n * 65 + k0 + khalf];
      b.y = sW[n * 65 + k0 + khalf + 1];
      acc[nt] = __builtin_amdgcn_wmma_f32_16x16x4_f32(
          /*neg_a=*/false, a, /*neg_b=*/false, b,
          /*c_mod=*/(short)0, acc[nt], /*reuse_a=*/false, /*reuse_b=*/false);
    }
  }

  const int nlane = lane & 15;
  const int mAdd  = (lane >> 4) * 8;
#pragma unroll
  for (int nt = 0; nt < NTILES; ++nt) {
    int o = nt * 16 + nlane;
    float sc = s[o], bi = bia[o];
#pragma unroll
    for (int v = 0; v < 8; ++v) {
      size_t m  = mBase + (size_t)(v + mAdd);
      float val = sc * acc[nt][v] + bi;
      if (doRelu) val = val > 0.0f ? val : 0.0f;
      size_t oi = m * CC + (size_t)(outOffset + o);
      if (skip) val += skip[oi];
      out[oi] = val;
    }
  }
}

// ---------------------------------------------------------------------------
// Kernel 4: edge = relu(cs*dot(em,cw)+cb);  out = cs*(dot_c cw*x + edge*sum(cw)) + cb
// ---------------------------------------------------------------------------
__global__ __launch_bounds__(256) void final_kernel(
    const float* __restrict__ x, const float* __restrict__ em,
    const float* __restrict__ cw, const float* __restrict__ cs_p,
    const float* __restrict__ cb_p, float* __restrict__ out) {
  int p  = blockIdx.x * 256 + threadIdx.x;   // 0 .. B*HW-1
  int b  = p >> 14;
  int hw = p & (HW - 1);
  const float cs = cs_p[0], cb = cb_p[0];
  const float* emp = em + (size_t)p * CC;
  const float* xp  = x + (size_t)b * CC * HW + hw;
  float de = 0.f, dx = 0.f, sw = 0.f;
#pragma unroll 8
  for (int c = 0; c < CC; ++c) {
    float wv = cw[c];
    de += wv * emp[c];
    sw += wv;
    dx += wv * xp[(size_t)c * HW];
  }
  float edge = cs * de + cb;
  edge = edge > 0.0f ? edge : 0.0f;
  out[p] = cs * (dx + edge * sw) + cb;
}

// ---------------------------------------------------------------------------
// Host launcher
// ---------------------------------------------------------------------------
extern "C" void kernel_launch(void* const* d_in, const int* in_sizes, int n_in,
                              void* d_out, int out_size, void* d_ws, size_t ws_size,
                              hipStream_t stream) {
  (void)in_sizes; (void)n_in; (void)out_size; (void)ws_size;

  const float* x      = (const float*)d_in[0];
  const float* dws_dw = (const float*)d_in[1];
  const float* dws_s1 = (const float*)d_in[2];
  const float* dws_b1 = (const float*)d_in[3];
  const float* dws_pw = (const float*)d_in[4];
  const float* dws_s2 = (const float*)d_in[5];
  const float* dws_b2 = (const float*)d_in[6];
  // branches: dw_w, dw_s, dw_b, pw_w, pw_s, pw_b at 7+6*i ...
  const float* conv_w = (const float*)d_in[31];
  const float* conv_s = (const float*)d_in[32];
  const float* conv_b = (const float*)d_in[33];
  const int*   radius = (const int*)d_in[34];

  float* ws = (float*)d_ws;
  float* xh = ws;             // [M][C] channel-last, 128 MiB
  float* t  = ws + NTOT;      // depthwise temp (reused by all branches)
  float* e  = ws + 2 * NTOT;  // DWS output / skip
  float* em = ws + 3 * NTOT;  // concat + skip accumulator

  // Allow >64KB dynamic LDS for the FFT kernel (CDNA5 WGP has 320KB).
  hipFuncSetAttribute((const void*)fft_highpass_kernel,
                      hipFuncAttributeMaxDynamicSharedMemorySize, FFT_LDS_BYTES);

  // 1) FFT high-pass magnitude -> xh (channel-last)
  fft_highpass_kernel<<<BB * CC, 256, FFT_LDS_BYTES, stream>>>(x, xh, radius);

  const int dwBlocks = (int)(NTOT / 256);
  const int pwBlocks = (int)(MTOT / 128);

  // 2) DWSConv: depthwise 3x3 + BN + ReLU, then pointwise 64->64 + BN + ReLU
  dw_aff_relu_kernel<<<dwBlocks, 256, 0, stream>>>(xh, dws_dw, dws_s1, dws_b1, t, 3, 1, 1);
  pw_wmma_kernel<4><<<pwBlocks, 256, 0, stream>>>(t, dws_pw, dws_s2, dws_b2,
                                                  nullptr, e, 0, /*relu=*/1);

  // 3) Four branches: depthwise(+BN+ReLU) then pointwise 64->16 (+BN), skip-add e
  const int ksz[4] = {1, 3, 3, 3};
  const int dil[4] = {1, 1, 3, 5};
  const int pad[4] = {0, 1, 3, 5};
  for (int i = 0; i < 4; ++i) {
    const float* dw_w = (const float*)d_in[7 + 6 * i + 0];
    const float* dw_s = (const float*)d_in[7 + 6 * i + 1];
    const float* dw_b = (const float*)d_in[7 + 6 * i + 2];
    const float* pw_w = (const float*)d_in[7 + 6 * i + 3];
    const float* pw_s = (const float*)d_in[7 + 6 * i + 4];
    const float* pw_b = (const float*)d_in[7 + 6 * i + 5];
    dw_aff_relu_kernel<<<dwBlocks, 256, 0, stream>>>(e, dw_w, dw_s, dw_b, t,
                                                     ksz[i], dil[i], pad[i]);
    pw_wmma_kernel<1><<<pwBlocks, 256, 0, stream>>>(t, pw_w, pw_s, pw_b,
                                                    /*skip=*/e, em, 16 * i, /*relu=*/0);
  }

  // 4) edge + final shared pointwise (folded)
  final_kernel<<<(int)(MTOT / 256), 256, 0, stream>>>(x, em, conv_w, conv_s, conv_b,
                                                      (float*)d_out);
}